// FilterInterpolationModule_15393162789086
// MI455X (gfx1250) — compile-verified
//
#include <hip/hip_runtime.h>

// FilterInterpolation (DAIN-style): B=4, C=3, H=544, W=960, 4x4 adaptive filter.
// HBM-bound (~200 MB @ 23.3 TB/s => ~8.6 us floor). No matmul structure -> no WMMA;
// instead use CDNA5 async global->LDS staging + NT cache hints + wave32 sizing.

#define TILE_X 256

#if __has_builtin(__builtin_amdgcn_global_load_async_to_lds_b32)
#define HAVE_ASYNC_LDS 1
#endif

#ifdef HAVE_ASYNC_LDS
// Builtin signature (per hipcc diagnostic): (__device__ int*, __shared__ int*, imm, imm)
#define AS1CAST(p) ((__attribute__((address_space(1))) int*)(p))
#define AS3CAST(p) ((__attribute__((address_space(3))) int*)(p))
#endif

__global__ __launch_bounds__(TILE_X)
void filter_interp_kernel(const float* __restrict__ img,   // [B,C,H,W]
                          const float* __restrict__ flow,  // [B,2,H,W]
                          const float* __restrict__ filt,  // [B,16,H,W]
                          float* __restrict__ out,         // [B,C,H,W]
                          int B, int C, int H, int W)
{
    const int tid = threadIdx.x;
    const int x   = blockIdx.x * TILE_X + tid;
    const int y   = blockIdx.y;
    const int b   = blockIdx.z;
    const bool active = (x < W);
    const int hw  = H * W;
    const int pix = y * W + (active ? x : (W - 1));   // clamp for safe addressing

    // ---- flow (streamed once: non-temporal) ----
    const float fx = __builtin_nontemporal_load(&flow[(b * 2    ) * hw + pix]);
    const float fy = __builtin_nontemporal_load(&flow[(b * 2 + 1) * hw + pix]);

    // ---- warp coords, mask, bilinear coefficients ----
    const float x2 = (float)x + fx;
    const float y2 = (float)y + fy;
    const bool m = (x2 >= 0.0f) & (y2 >= 0.0f) &
                   (x2 <= (float)(W - 1)) & (y2 <= (float)(H - 1)) &
                   (fabsf(fx) < 0.5f * (float)W) & (fabsf(fy) < 0.5f * (float)H);

    if (!active || !m) {
        if (active) {
            for (int c = 0; c < C; ++c)
                __builtin_nontemporal_store(0.0f, &out[(b * C + c) * hw + pix]);
        }
        return;
    }

    const float x2c = fminf(fmaxf(x2, 0.0f), (float)(W - 1));
    const float y2c = fminf(fmaxf(y2, 0.0f), (float)(H - 1));
    const int ix = (int)floorf(x2c);
    const int iy = (int)floorf(y2c);
    const float alpha = x2c - (float)ix;
    const float beta  = y2c - (float)iy;
    const float wa = (1.0f - alpha) * (1.0f - beta);
    const float wb = alpha * (1.0f - beta);
    const float wc = (1.0f - alpha) * beta;
    const float wd = alpha * beta;

    // 5x5 footprint indices (clipped). ixL = ix-1, iyT = iy-1 for fs=4.
    int cols[5], rows[5];
    #pragma unroll
    for (int i = 0; i < 5; ++i) {
        cols[i] = min(max(ix - 1 + i, 0), W - 1);
        rows[i] = min(max(iy - 1 + i, 0), H - 1);
    }

    // ---- 16 adaptive filter weights ----
    float w3[16];
#ifdef HAVE_ASYNC_LDS
    // Stage input3 through LDS via CDNA5 async global->LDS DMA path.
    // Each lane writes & reads only its own slot, so only ASYNCcnt drain is
    // needed (no workgroup barrier -> no divergence hazard from early returns).
    __shared__ float s_w[16][TILE_X];
    {
        const float* gbase = filt + b * 16 * hw + pix;
        #pragma unroll
        for (int t = 0; t < 16; ++t) {
            __builtin_amdgcn_global_load_async_to_lds_b32(
                AS1CAST(gbase + t * hw), AS3CAST(&s_w[t][tid]), 0, 0);
        }
#if __has_builtin(__builtin_amdgcn_s_wait_asynccnt)
        __builtin_amdgcn_s_wait_asynccnt(0);
#else
        asm volatile("s_wait_asynccnt 0" ::: "memory");
#endif
        #pragma unroll
        for (int t = 0; t < 16; ++t) w3[t] = s_w[t][tid];
    }
#else
    #pragma unroll
    for (int t = 0; t < 16; ++t)
        w3[t] = __builtin_nontemporal_load(&filt[(b * 16 + t) * hw + pix]);
#endif

    // ---- per channel: gather 5x5 patch, 4 correlations, bilinear combine ----
    #pragma unroll 3
    for (int c = 0; c < C; ++c) {
        const float* imc = img + (b * C + c) * hw;
        float S00 = 0.0f, S01 = 0.0f, S10 = 0.0f, S11 = 0.0f;
        #pragma unroll
        for (int j = 0; j < 5; ++j) {
            const float* rp = imc + rows[j] * W;
            float p[5];
            #pragma unroll
            for (int i = 0; i < 5; ++i) p[i] = rp[cols[i]];   // temporal: keep in L2
            if (j < 4) {            // acts as "top" row for filter row j
                #pragma unroll
                for (int i = 0; i < 4; ++i) {
                    const float w = w3[4 * j + i];
                    S00 = fmaf(w, p[i],     S00);
                    S01 = fmaf(w, p[i + 1], S01);
                }
            }
            if (j > 0) {            // acts as "bottom" row for filter row j-1
                #pragma unroll
                for (int i = 0; i < 4; ++i) {
                    const float w = w3[4 * (j - 1) + i];
                    S10 = fmaf(w, p[i],     S10);
                    S11 = fmaf(w, p[i + 1], S11);
                }
            }
        }
        const float val = fmaf(wa, S00, fmaf(wb, S01, fmaf(wc, S10, wd * S11)));
        __builtin_nontemporal_store(val, &out[(b * C + c) * hw + pix]);
    }
}

extern "C" void kernel_launch(void* const* d_in, const int* in_sizes, int n_in,
                              void* d_out, int out_size, void* d_ws, size_t ws_size,
                              hipStream_t stream) {
    const float* img  = (const float*)d_in[0];
    const float* flow = (const float*)d_in[1];
    const float* filt = (const float*)d_in[2];
    float* out = (float*)d_out;

    const int H = 544, W = 960;
    const int B = in_sizes[1] / (2 * H * W);       // = 4
    const int C = in_sizes[0] / (B * H * W);       // = 3

    dim3 block(TILE_X, 1, 1);
    dim3 grid((W + TILE_X - 1) / TILE_X, H, B);
    filter_interp_kernel<<<grid, block, 0, stream>>>(img, flow, filt, out, B, C, H, W);
}